// CenterLoss_79723182948888
// MI455X (gfx1250) — compile-verified
//
#include <hip/hip_runtime.h>

// CenterLoss forward: mean((x - centers[labels])^2)
// Memory-bound streaming reduction (~256 MB @ 23.3 TB/s ≈ 11 us).
// Wave-per-row layout: uniform label load, coalesced gather, NT loads on the
// one-shot input stream, f32 WMMA (16x16x4, B=ones) for the wave reduction.

typedef __attribute__((ext_vector_type(2))) float v2f;
typedef __attribute__((ext_vector_type(4))) float v4f;
typedef __attribute__((ext_vector_type(8))) float v8f;

#define FEAT 512
#define BLOCK_THREADS 256
#define WAVES_PER_BLOCK (BLOCK_THREADS / 32)
#define GRID_BLOCKS 2048

__global__ __launch_bounds__(BLOCK_THREADS)
void center_loss_partial(const float* __restrict__ x,
                         const int* __restrict__ labels,
                         const float* __restrict__ centers,
                         float* __restrict__ partials,
                         int rows)
{
    const int lane = threadIdx.x & 31;
    const int wave = threadIdx.x >> 5;
    const int globalWave = blockIdx.x * WAVES_PER_BLOCK + wave;
    const int totalWaves = gridDim.x * WAVES_PER_BLOCK;

    float acc0 = 0.0f, acc1 = 0.0f;

    for (int r = globalWave; r < rows; r += totalWaves) {
        const int lbl = labels[r];                   // wave-uniform
        const float* xr = x + (size_t)r * FEAT;
        const float* cr = centers + (size_t)lbl * FEAT;

        // Prefetch next row of the input stream (global_prefetch_b8).
        if (r + totalWaves < rows) {
            __builtin_prefetch(x + (size_t)(r + totalWaves) * FEAT + lane * 4, 0, 1);
        }

        #pragma unroll
        for (int j = 0; j < FEAT / 128; ++j) {       // 4 chunks of 128 floats
            const int f = j * 128 + lane * 4;
            v4f xv = __builtin_nontemporal_load((const v4f*)(xr + f)); // read-once stream
            v4f cv = *(const v4f*)(cr + f);                            // L2-hot gather
            v4f d  = xv - cv;
            acc0 = fmaf(d.x, d.x, acc0);
            acc1 = fmaf(d.y, d.y, acc1);
            acc0 = fmaf(d.z, d.z, acc0);
            acc1 = fmaf(d.w, d.w, acc1);
        }
    }

    // Wave reduction via one V_WMMA_F32_16X16X4_F32 with B = ones:
    // D[m][n] = sum_k A[m][k], i.e. row sums of the 64 per-lane partials.
    // Per the CDNA5 C/D layout, summing the 8 D VGPRs gives lanes 0-15 the
    // total of rows 0-7 and lanes 16-31 the total of rows 8-15.
    v2f a; a.x = acc0; a.y = acc1;
    v2f b; b.x = 1.0f; b.y = 1.0f;
    v8f c = {};
    v8f dm = __builtin_amdgcn_wmma_f32_16x16x4_f32(
        /*neg_a=*/false, a, /*neg_b=*/false, b,
        /*c_mod=*/(short)0, c, /*reuse_a=*/false, /*reuse_b=*/false);

    float t = dm[0] + dm[1] + dm[2] + dm[3] + dm[4] + dm[5] + dm[6] + dm[7];
    t += __shfl_xor(t, 16, 32);                      // wave total in every lane

    __shared__ float sdata[WAVES_PER_BLOCK];
    if (lane == 0) sdata[wave] = t;
    __syncthreads();

    if (threadIdx.x == 0) {
        float s = 0.0f;
        #pragma unroll
        for (int w = 0; w < WAVES_PER_BLOCK; ++w) s += sdata[w];
        partials[blockIdx.x] = s;                    // deterministic per-block partial
    }
}

__global__ __launch_bounds__(BLOCK_THREADS)
void center_loss_final(const float* __restrict__ partials, int n,
                       float* __restrict__ out, float scale)
{
    __shared__ float s[BLOCK_THREADS];
    float t = 0.0f;
    for (int i = threadIdx.x; i < n; i += BLOCK_THREADS) t += partials[i];
    s[threadIdx.x] = t;
    __syncthreads();
    #pragma unroll
    for (int w = BLOCK_THREADS / 2; w > 0; w >>= 1) {
        if (threadIdx.x < w) s[threadIdx.x] += s[threadIdx.x + w];
        __syncthreads();
    }
    if (threadIdx.x == 0) out[0] = s[0] * scale;
}

extern "C" void kernel_launch(void* const* d_in, const int* in_sizes, int n_in,
                              void* d_out, int out_size, void* d_ws, size_t ws_size,
                              hipStream_t stream)
{
    const float* x       = (const float*)d_in[0];   // [N, 512] f32
    const int*   labels  = (const int*)d_in[1];     // [N] i32
    const float* centers = (const float*)d_in[2];   // [1000, 512] f32
    float* out      = (float*)d_out;                // scalar f32
    float* partials = (float*)d_ws;                 // GRID_BLOCKS floats

    const int rows = in_sizes[1];                   // N
    center_loss_partial<<<GRID_BLOCKS, BLOCK_THREADS, 0, stream>>>(
        x, labels, centers, partials, rows);

    const float scale = (float)(1.0 / ((double)rows * (double)FEAT));
    center_loss_final<<<1, BLOCK_THREADS, 0, stream>>>(
        partials, GRID_BLOCKS, out, scale);
}